// Embedder_Transformer_81466939670847
// MI455X (gfx1250) — compile-verified
//
#include <hip/hip_runtime.h>
#include <hip/hip_bf16.h>
#include <math.h>

// ---------------- problem constants (from reference) ----------------
#define N_NODES 50000
#define N_EDGES 800000
#define DIM     96
#define N_LAYERS 3
#define N_GRAPHS 512
#define SCALE_F 0.10206207261596577f   // 1/sqrt(96)

typedef __attribute__((ext_vector_type(2))) float v2f;
typedef __attribute__((ext_vector_type(8))) float v8f;

// encoded -inf for ordered-uint float max
#define ENC_NEG_INF 0x007FFFFFu

// LDS row stride (floats) for the staged weight matrix: 100 -> bank stride 36,
// gcd(36,64)=4 so the 16 B-fragment lanes hit 16 distinct banks, and the upper
// half-wave (+2 floats) lands on the disjoint (== 2 mod 4) bank set.
#define LDS_W_STRIDE 100

// ======================================================================
// Fused GEMM: out = h @ W.T + bias for one of {Wq,Wk,Wv,Ws} per block.y.
// Block = 256 threads (8 waves). The block stages the full 96x96 weight
// matrix (36 KB) + bias into LDS; each wave then computes a 16x96 output
// strip: 6 f32 accumulators, A fragment loaded once per K-step and reused
// across the 6 column tiles (144 V_WMMA_F32_16X16X4_F32 per wave).
// ======================================================================
__global__ __launch_bounds__(256)
void gemm_qkvs_kernel(const float* __restrict__ h,
                      const float* __restrict__ Wq, const float* __restrict__ bq,
                      const float* __restrict__ Wk, const float* __restrict__ bk,
                      const float* __restrict__ Wv, const float* __restrict__ bv,
                      const float* __restrict__ Ws, const float* __restrict__ bs,
                      float* __restrict__ q, float* __restrict__ k,
                      float* __restrict__ v, float* __restrict__ hout)
{
    __shared__ float sW[DIM * LDS_W_STRIDE];   // 38,400 B
    __shared__ float sBias[DIM];

    const float* W; const float* bias; float* out;
    switch (blockIdx.y) {
        case 0:  W = Wq; bias = bq; out = q;    break;
        case 1:  W = Wk; bias = bk; out = k;    break;
        case 2:  W = Wv; bias = bv; out = v;    break;
        default: W = Ws; bias = bs; out = hout; break;  // skip path -> h_out
    }

    // cooperative stage: 9216 floats by 256 threads (36 each)
    for (int i = threadIdx.x; i < DIM * DIM; i += 256) {
        const int r = i / DIM, c = i % DIM;
        sW[r * LDS_W_STRIDE + c] = W[i];
    }
    if (threadIdx.x < DIM) sBias[threadIdx.x] = bias[threadIdx.x];
    __syncthreads();

    const int wave    = threadIdx.x >> 5;
    const int lane    = threadIdx.x & 31;
    const int rowBase = blockIdx.x * 128 + wave * 16;
    if (rowBase >= N_NODES) return;      // wave-uniform: EXEC stays all-ones

    const int n16  = lane & 15;
    const int half = lane >> 4;          // selects K pair {0,1} vs {2,3}
    // A frag: lane (m=n16) holds h[rowBase+m][k0 + 2*half + {0,1}]
    const float* aRow = h + (size_t)(rowBase + n16) * DIM + 2 * half;

    v8f acc[6] = {};
    for (int k0 = 0; k0 < DIM; k0 += 4) {
        const v2f a = *(const v2f*)(aRow + k0);
        #pragma unroll
        for (int t = 0; t < 6; ++t) {
            // B frag: B[k][n] = W[t*16+n][k], served from LDS
            const v2f b = *(const v2f*)(&sW[(t * 16 + n16) * LDS_W_STRIDE + k0 + 2 * half]);
            acc[t] = __builtin_amdgcn_wmma_f32_16x16x4_f32(
                         /*neg_a=*/false, a, /*neg_b=*/false, b,
                         /*c_mod=*/(short)0, acc[t],
                         /*reuse_a=*/false, /*reuse_b=*/false);
        }
    }

    #pragma unroll
    for (int t = 0; t < 6; ++t) {
        const int colBase = t * 16;
        const float bval = sBias[colBase + n16];
        // D layout: VGPR r -> (M = r + 8*half, N = n16)
        #pragma unroll
        for (int r = 0; r < 8; ++r) {
            const int row = rowBase + r + 8 * half;
            out[(size_t)row * DIM + colBase + n16] = acc[t][r] + bval;
        }
    }
}

// ---------------- per-layer node scratch init ----------------
__global__ void init_nodes_kernel(unsigned* __restrict__ mEnc,
                                  float* __restrict__ denom)
{
    int n = blockIdx.x * blockDim.x + threadIdx.x;
    if (n < N_NODES) { mEnc[n] = ENC_NEG_INF; denom[n] = 0.0f; }
}

// ---------------- edge logits + segment max (one wave per edge) ----------------
__global__ __launch_bounds__(256)
void logits_kernel(const float* __restrict__ q, const float* __restrict__ k,
                   const long long* __restrict__ src, const long long* __restrict__ dst,
                   float* __restrict__ logits, unsigned* __restrict__ mEnc)
{
    const int e    = (blockIdx.x * blockDim.x + threadIdx.x) >> 5;
    const int lane = threadIdx.x & 31;
    if (e >= N_EDGES) return;
    const long long s = src[e], d = dst[e];
    const float* qr = q + (size_t)d * DIM;
    const float* kr = k + (size_t)s * DIM;
    float acc = 0.0f;
    #pragma unroll
    for (int i = 0; i < 3; ++i) acc += qr[lane + 32 * i] * kr[lane + 32 * i];
    #pragma unroll
    for (int off = 16; off > 0; off >>= 1) acc += __shfl_xor(acc, off, 32);
    if (lane == 0) {
        acc *= SCALE_F;
        logits[e] = acc;
        unsigned u = __float_as_uint(acc);
        u = (u & 0x80000000u) ? ~u : (u | 0x80000000u);   // order-preserving encode
        atomicMax(mEnc + d, u);
    }
}

// ---------------- exp + segment-sum of exp (one thread per edge) ----------------
// Every dst here has >= 1 incoming edge, so its segment max is finite;
// the reference's isolated-node guard only affects nodes with no edges.
__global__ void exp_kernel(float* __restrict__ logits,
                           const long long* __restrict__ dst,
                           const unsigned* __restrict__ mEnc,
                           float* __restrict__ denom)
{
    int e = blockIdx.x * blockDim.x + threadIdx.x;
    if (e < N_EDGES) {
        const long long d = dst[e];
        const unsigned u = mEnc[d];
        const unsigned bbits = (u & 0x80000000u) ? (u & 0x7FFFFFFFu) : ~u;
        const float m = __uint_as_float(bbits);
        const float ex = __expf(logits[e] - m);
        logits[e] = ex;                    // reuse buffer for exp(logit - m)
        atomicAdd(denom + d, ex);
    }
}

// ---------------- weighted scatter-add of values (one wave per edge) ----------------
__global__ __launch_bounds__(256)
void agg_kernel(const float* __restrict__ expe, const float* __restrict__ denom,
                const float* __restrict__ v,
                const long long* __restrict__ src, const long long* __restrict__ dst,
                float* __restrict__ hout)
{
    const int e    = (blockIdx.x * blockDim.x + threadIdx.x) >> 5;
    const int lane = threadIdx.x & 31;
    if (e >= N_EDGES) return;
    const long long s = src[e], d = dst[e];
    const float alpha = expe[e] / fmaxf(denom[d], 1e-16f);
    const float* vr = v + (size_t)s * DIM;
    float* o = hout + (size_t)d * DIM;
    #pragma unroll
    for (int i = 0; i < 3; ++i) {
        const int idx = lane + 32 * i;
        atomicAdd(o + idx, alpha * vr[idx]);
    }
}

// ---------------- ReLU in place ----------------
__global__ void relu_kernel(float* __restrict__ h)
{
    int t = blockIdx.x * blockDim.x + threadIdx.x;
    if (t < N_NODES * DIM) h[t] = fmaxf(h[t], 0.0f);
}

// ---------------- mean pool over graphs ----------------
__global__ void pool_zero_kernel(float* __restrict__ pool, float* __restrict__ cnt)
{
    int t = blockIdx.x * blockDim.x + threadIdx.x;
    if (t < N_GRAPHS * DIM) pool[t] = 0.0f;
    if (t < N_GRAPHS)       cnt[t]  = 0.0f;
}

__global__ void pool_acc_kernel(const float* __restrict__ h,
                                const long long* __restrict__ batch,
                                float* __restrict__ pool, float* __restrict__ cnt)
{
    int t = blockIdx.x * blockDim.x + threadIdx.x;
    if (t < N_NODES * DIM) {
        const int n = t / DIM, dloc = t % DIM;
        const long long g = batch[n];
        atomicAdd(pool + (size_t)g * DIM + dloc, h[t]);
        if (dloc == 0) atomicAdd(cnt + g, 1.0f);
    }
}

__global__ void pool_div_kernel(const float* __restrict__ pool,
                                const float* __restrict__ cnt,
                                float* __restrict__ out)
{
    int t = blockIdx.x * blockDim.x + threadIdx.x;
    if (t < N_GRAPHS * DIM) out[t] = pool[t] / fmaxf(cnt[t / DIM], 1.0f);
}

// ======================================================================
extern "C" void kernel_launch(void* const* d_in, const int* in_sizes, int n_in,
                              void* d_out, int out_size, void* d_ws, size_t ws_size,
                              hipStream_t stream)
{
    (void)in_sizes; (void)n_in; (void)out_size; (void)ws_size;

    // setup_inputs() order: x, Wq, bq, Wk, bk, Wv, bv, Ws, bs, edge_index, batch
    const float*     x   = (const float*)d_in[0];
    const float*     Wq  = (const float*)d_in[1];
    const float*     bq  = (const float*)d_in[2];
    const float*     Wk  = (const float*)d_in[3];
    const float*     bk  = (const float*)d_in[4];
    const float*     Wv  = (const float*)d_in[5];
    const float*     bv  = (const float*)d_in[6];
    const float*     Ws  = (const float*)d_in[7];
    const float*     bs  = (const float*)d_in[8];
    const long long* ei  = (const long long*)d_in[9];   // int64 [2, E]
    const long long* bat = (const long long*)d_in[10];  // int64 [N]
    const long long* src = ei;
    const long long* dst = ei + N_EDGES;

    // ---- workspace carve-out ----
    char* ws = (char*)d_ws;
    size_t off = 0;
    auto carve = [&](size_t bytes) -> void* {
        void* p = ws + off;
        off = (off + bytes + 255) & ~(size_t)255;
        return p;
    };
    const size_t ND = (size_t)N_NODES * DIM * sizeof(float);
    float*    q      = (float*)   carve(ND);
    float*    k      = (float*)   carve(ND);
    float*    v      = (float*)   carve(ND);
    float*    hA     = (float*)   carve(ND);
    float*    hB     = (float*)   carve(ND);
    float*    logit  = (float*)   carve((size_t)N_EDGES * sizeof(float));
    unsigned* mEnc   = (unsigned*)carve((size_t)N_NODES * sizeof(unsigned));
    float*    denom  = (float*)   carve((size_t)N_NODES * sizeof(float));
    float*    pool   = (float*)   carve((size_t)N_GRAPHS * DIM * sizeof(float));
    float*    cnt    = (float*)   carve((size_t)N_GRAPHS * sizeof(float));

    const int TB = 256;
    const dim3 gGemm((N_NODES + 127) / 128, 4);           // 8 waves x 16 rows, 4 mats
    const int  gNode  = (N_NODES + TB - 1) / TB;
    const int  gEdgeW = ((N_EDGES * 32) + TB - 1) / TB;   // wave-per-edge kernels
    const int  gEdgeT = (N_EDGES + TB - 1) / TB;
    const int  gND    = (N_NODES * DIM + TB - 1) / TB;
    const int  gGD    = (N_GRAPHS * DIM + TB - 1) / TB;

    const float* hin = x;
    float* houts[N_LAYERS] = { hA, hB, hA };

    for (int l = 0; l < N_LAYERS; ++l) {
        float* hout = houts[l];
        const size_t wOff = (size_t)l * DIM * DIM;
        const size_t bOff = (size_t)l * DIM;

        init_nodes_kernel<<<gNode, TB, 0, stream>>>(mEnc, denom);

        gemm_qkvs_kernel<<<gGemm, 256, 0, stream>>>(
            hin,
            Wq + wOff, bq + bOff, Wk + wOff, bk + bOff,
            Wv + wOff, bv + bOff, Ws + wOff, bs + bOff,
            q, k, v, hout);

        logits_kernel<<<gEdgeW, TB, 0, stream>>>(q, k, src, dst, logit, mEnc);
        exp_kernel<<<gEdgeT, TB, 0, stream>>>(logit, dst, mEnc, denom);
        agg_kernel<<<gEdgeW, TB, 0, stream>>>(logit, denom, v, src, dst, hout);

        if (l < N_LAYERS - 1)
            relu_kernel<<<gND, TB, 0, stream>>>(hout);

        hin = hout;
    }

    pool_zero_kernel<<<gGD, TB, 0, stream>>>(pool, cnt);
    pool_acc_kernel<<<gND, TB, 0, stream>>>(hin, bat, pool, cnt);
    pool_div_kernel<<<gGD, TB, 0, stream>>>(pool, cnt, (float*)d_out);
}